// ButterflyPermuter_61942018343003
// MI455X (gfx1250) — compile-verified
//
#include <hip/hip_runtime.h>
#include <stdint.h>

#define DIM 1024
#define NUM_STAGES 10
#define TAB_ENTRIES (NUM_STAGES * 512)      /* 5120 float2 = 40 KB */
#define WAVES_PER_BLOCK 8
#define TOKENS_PER_WAVE 8
#define TOKENS_PER_BLOCK (WAVES_PER_BLOCK * TOKENS_PER_WAVE)

typedef unsigned int v4u __attribute__((ext_vector_type(4)));
typedef int          v4i __attribute__((ext_vector_type(4)));
typedef int          v8i __attribute__((ext_vector_type(8)));

#if defined(__HIP_DEVICE_COMPILE__) && defined(__gfx1250__) && \
    __has_builtin(__builtin_amdgcn_tensor_load_to_lds)
#define HAVE_TDM 1
#else
#define HAVE_TDM 0
#endif

// ---------------------------------------------------------------------------
// Kernel 1: evaluate cos/sin once for all 10*512 angles and store them in a
// per-stage bank-conflict-free layout:
//   stages 0..4 : slot = c*32 + L   holds theta_idx = 16*L + c   (in-lane stage)
//   stages 5..9 : slot = k*16 + H   holds theta_idx = 32*H + k   (cross-lane stage)
// ---------------------------------------------------------------------------
__global__ void butterfly_build_tab(const float* __restrict__ angles,
                                    float2* __restrict__ gtab) {
  int t = blockIdx.x * blockDim.x + threadIdx.x;
  if (t >= TAB_ENTRIES) return;
  int s = t >> 9;
  int slot = t & 511;
  int theta_idx;
  if (s < 5) {
    int c = slot >> 5, L = slot & 31;
    theta_idx = 16 * L + c;
  } else {
    int k = slot >> 4, H = slot & 15;
    theta_idx = 32 * H + k;
  }
  float a = angles[(s << 9) + theta_idx];
  float sn, cn;
  sincosf(a, &sn, &cn);
  gtab[t] = make_float2(cn, sn);
}

// ---------------------------------------------------------------------------
// Kernel 2: one wave32 per token; lane L holds elements [32L, 32L+32).
// Stages 0..4 in-lane, stages 5..9 via __shfl_xor. cos/sin table in LDS,
// filled once per block by the Tensor Data Mover.
// ---------------------------------------------------------------------------
__global__ __launch_bounds__(256) void butterfly_apply(
    const float* __restrict__ x, const float2* __restrict__ gtab,
    float* __restrict__ out, int nTokens) {
  __shared__ __align__(16) float2 tab[TAB_ENTRIES];

  const int lane = threadIdx.x & 31;
  const int wave = threadIdx.x >> 5;

  // Compiler-visible (never-executed) store: guarantees the LDS array has a
  // store in the module so loads cannot be folded to undef when the real fill
  // is done by the opaque TDM engine.
  if (nTokens < 0) tab[0] = make_float2(0.0f, 0.0f);

#if HAVE_TDM
  if (wave == 0) {
    // Tensor DMA descriptor (ISA cdna5 §8): 1-D tile, 10240 dwords, LDS offset 0.
    uint64_t ga = (uint64_t)(const void*)gtab;
    v4u g0 = { 1u,                                   // count=1 (valid), no gather
               0u,                                   // lds_addr = 0 (tab @ LDS base)
               (uint32_t)ga,                         // global_addr[31:0]
               (uint32_t)((ga >> 32) & 0x01FFFFFFu)  // global_addr[56:32]
                   | (2u << 30) };                   // type = 2 ("image")
    const uint32_t n = TAB_ENTRIES * 2;              // 10240 elements of 4B
    v8i g1 = { (int)(2u << 16),                      // wg_mask=0, data_size=2 (4B)
               (int)((n & 0xFFFFu) << 16),           // tensor_dim0[15:0] @ bits 63:48
               (int)(1u << 16),                      // tensor_dim0 hi=0, tensor_dim1=1
               (int)((n & 0xFFFFu) << 16),           // tile_dim0 @ bits 127:112
               1,                                    // tile_dim1=1, tile_dim2=0
               (int)n,                               // tensor_dim0_stride[31:0]
               0, 0 };
    v4i gz = {0, 0, 0, 0};
#if __clang_major__ >= 23
    v8i gz8 = {0, 0, 0, 0, 0, 0, 0, 0};
    __builtin_amdgcn_tensor_load_to_lds(g0, g1, gz, gz, gz8, 0);
#else
    __builtin_amdgcn_tensor_load_to_lds(g0, g1, gz, gz, 0);
#endif
    __builtin_amdgcn_s_wait_tensorcnt(0);
  }
#else
  {
    const float4* src = (const float4*)gtab;
    float4* dst = (float4*)tab;
    for (int i = threadIdx.x; i < TAB_ENTRIES / 2; i += blockDim.x)
      dst[i] = src[i];
  }
#endif
  __syncthreads();

  // Escape the LDS array's address into opaque asm with a memory clobber:
  // the optimizer must now assume its contents were written (by the TDM).
  {
    uintptr_t tab_addr = (uintptr_t)&tab[0];
    asm volatile("" : "+v"(tab_addr) : : "memory");
  }

  const int tokenBase = blockIdx.x * TOKENS_PER_BLOCK + wave * TOKENS_PER_WAVE;

#pragma unroll 1
  for (int ti = 0; ti < TOKENS_PER_WAVE; ++ti) {
    const int token = tokenBase + ti;
    if (token >= nTokens) break;

    // Prefetch next token's slice of this lane while we compute this one.
    if (token + 1 < nTokens)
      __builtin_prefetch(x + (size_t)(token + 1) * DIM + lane * 32, 0, 1);

    float v[32];
    const float4* xin = (const float4*)(x + (size_t)token * DIM + lane * 32);
#pragma unroll
    for (int q = 0; q < 8; ++q) {
      float4 t4 = xin[q];
      v[4 * q + 0] = t4.x; v[4 * q + 1] = t4.y;
      v[4 * q + 2] = t4.z; v[4 * q + 3] = t4.w;
    }

    // ---- stages 0..4: pair distance 1..16, entirely inside the lane ----
#pragma unroll
    for (int s = 0; s < 5; ++s) {
      const int half = 1 << s;
      const float2* st = tab + (s << 9);
#pragma unroll
      for (int c = 0; c < 16; ++c) {
        const int kl = ((c >> s) << (s + 1)) | (c & (half - 1)); // insert 0 at bit s
        const int kr = kl | half;
        const float2 cs = st[(c << 5) + lane];  // conflict-free: 64 banks hit once
        const float l = v[kl], r = v[kr];
        v[kl] = fmaf(cs.x, l, cs.y * r);
        v[kr] = fmaf(cs.x, r, -(cs.y * l));
      }
    }

    // ---- stages 5..9: pair distance 32..512 -> lane XOR 1,2,4,8,16 ----
#pragma unroll
    for (int s = 5; s < 10; ++s) {
      const int m = 1 << (s - 5);
      const float2* st = tab + (s << 9);
      const int H = ((lane >> (s - 4)) << (s - 5)) | (lane & (m - 1)); // drop bit s-5
      const float sgn = (lane & m) ? -1.0f : 1.0f;
#pragma unroll
      for (int k = 0; k < 32; ++k) {
        const float p = __shfl_xor(v[k], m, 32);        // partner's element
        const float2 cs = st[(k << 4) + H];             // 2-lane broadcast reads
        v[k] = fmaf(cs.x, v[k], sgn * (cs.y * p));
      }
    }

    float4* o = (float4*)(out + (size_t)token * DIM + lane * 32);
#pragma unroll
    for (int q = 0; q < 8; ++q)
      o[q] = make_float4(v[4 * q + 0], v[4 * q + 1], v[4 * q + 2], v[4 * q + 3]);
  }
}

// ---------------------------------------------------------------------------
extern "C" void kernel_launch(void* const* d_in, const int* in_sizes, int n_in,
                              void* d_out, int out_size, void* d_ws, size_t ws_size,
                              hipStream_t stream) {
  const float* x      = (const float*)d_in[0];
  const float* angles = (const float*)d_in[1];
  float* out          = (float*)d_out;
  float2* gtab        = (float2*)d_ws;              // 40 KB scratch table
  const int nTokens   = in_sizes[0] / DIM;

  butterfly_build_tab<<<(TAB_ENTRIES + 255) / 256, 256, 0, stream>>>(angles, gtab);

  const int blocks = (nTokens + TOKENS_PER_BLOCK - 1) / TOKENS_PER_BLOCK;
  butterfly_apply<<<blocks, 256, 0, stream>>>(x, gtab, out, nTokens);
}